// PointNetSetAbstraction_34634616275288
// MI455X (gfx1250) — compile-verified
//
#include <hip/hip_runtime.h>
#include <hip/hip_bf16.h>
#include <stdint.h>

// Problem constants (match reference)
#define BATCH   16
#define NSRC    8192               // input points per batch
#define NPTS    1024               // sampled points
#define GRP     32                 // group size
#define RAD2    0.04f              // 0.2^2
#define NTOT    (BATCH*NPTS*GRP)   // 524288 spatial positions for the MLP
#define NCHUNK  (NTOT/128)         // 4096 column chunks of 128
#define BN_EPS  1e-5f

typedef float v2f __attribute__((ext_vector_type(2)));
typedef float v8f __attribute__((ext_vector_type(8)));

// ---- gfx1250 async global->LDS path (guarded; falls back to sync copy) ----
#if defined(__has_builtin)
#  if __has_builtin(__builtin_amdgcn_global_load_async_to_lds_b128)
#    define USE_ASYNC_LDS 1
#  endif
#endif
#ifndef USE_ASYNC_LDS
#  define USE_ASYNC_LDS 0
#endif

// Signature learned from the round-2 diagnostic:
//   (addrspace(1) int4* src_global, addrspace(3) int4* dst_lds, i32 off, i32 cpol)
typedef int v4i __attribute__((vector_size(16)));
typedef __attribute__((address_space(1))) v4i* gv4i_p;
typedef __attribute__((address_space(3))) v4i* lv4i_p;

__device__ __forceinline__ void wait_async0() {
#if defined(__has_builtin) && __has_builtin(__builtin_amdgcn_s_wait_asynccnt)
  __builtin_amdgcn_s_wait_asynccnt(0);
#else
  asm volatile("s_wait_asynccnt 0" ::: "memory");
#endif
}

// ---------------------------------------------------------------------------
// Kernel 1: farthest point sampling. One workgroup per batch (16 WGPs busy).
// 8192 points + running min-dists live in registers (8/thread, 1024 threads).
// Writes new_xyz directly into d_out[0 .. 16*1024*3).
// ---------------------------------------------------------------------------
__global__ __launch_bounds__(1024) void fps_kernel(const float* __restrict__ xyz,
                                                   float* __restrict__ new_xyz) {
  const int b = blockIdx.x;
  const int t = threadIdx.x;
  __shared__ float sc[3];
  __shared__ float rv[32];
  __shared__ int   ri[32];
  __shared__ int   swin;

  float px[8], py[8], pz[8], dist[8];
  const float* base = xyz + (size_t)b * NSRC * 3;
#pragma unroll
  for (int k = 0; k < 8; ++k) {
    int p = k * 1024 + t;            // point p lives on thread p%1024, slot p/1024
    px[k] = base[p * 3 + 0];
    py[k] = base[p * 3 + 1];
    pz[k] = base[p * 3 + 2];
    dist[k] = 1e10f;
  }

  int farthest = 0;                  // reference: sample[0] == 0
  for (int j = 0; j < NPTS; ++j) {
    if (t == (farthest & 1023)) {    // owner broadcasts centroid coords
      int slot = farthest >> 10;
      sc[0] = px[slot]; sc[1] = py[slot]; sc[2] = pz[slot];
    }
    __syncthreads();
    float cx = sc[0], cy = sc[1], cz = sc[2];
    if (t == 0) {
      float* o = new_xyz + ((size_t)b * NPTS + j) * 3;
      o[0] = cx; o[1] = cy; o[2] = cz;
    }
    float bv = -1.0f; int bi = 0x7fffffff;
#pragma unroll
    for (int k = 0; k < 8; ++k) {
      float dx = px[k] - cx, dy = py[k] - cy, dz = pz[k] - cz;
      float d  = dx * dx + dy * dy + dz * dz;
      d = fminf(d, dist[k]);
      dist[k] = d;
      int p = k * 1024 + t;
      if (d > bv || (d == bv && p < bi)) { bv = d; bi = p; }
    }
    // wave32 argmax reduce (first-max index tie-break like jnp.argmax)
#pragma unroll
    for (int off = 16; off >= 1; off >>= 1) {
      float ov = __shfl_xor(bv, off, 32);
      int   oi = __shfl_xor(bi, off, 32);
      if (ov > bv || (ov == bv && oi < bi)) { bv = ov; bi = oi; }
    }
    if ((t & 31) == 0) { rv[t >> 5] = bv; ri[t >> 5] = bi; }
    __syncthreads();
    if (t < 32) {
      bv = rv[t]; bi = ri[t];
#pragma unroll
      for (int off = 16; off >= 1; off >>= 1) {
        float ov = __shfl_xor(bv, off, 32);
        int   oi = __shfl_xor(bi, off, 32);
        if (ov > bv || (ov == bv && oi < bi)) { bv = ov; bi = oi; }
      }
      if (t == 0) swin = bi;
    }
    __syncthreads();
    farthest = swin;
  }
}

// ---------------------------------------------------------------------------
// Kernel 2: ball query + gather + concat, emitting channel-major X0 [6][NTOT].
// One wave per query; ballot + popc-prefix collects the first 32 in-radius
// points in ascending index order (== sort-then-take-first-32 in reference),
// early-exiting once the group is full.
// ---------------------------------------------------------------------------
__global__ __launch_bounds__(256) void ballquery_kernel(const float* __restrict__ xyz,
                                                        const float* __restrict__ feat,
                                                        const float* __restrict__ new_xyz,
                                                        float* __restrict__ X0) {
  const int q    = blockIdx.x * 8 + (threadIdx.x >> 5);
  const int lane = threadIdx.x & 31;
  const int b = q >> 10;
  const int s = q & 1023;

  const float* ctr = new_xyz + ((size_t)b * NPTS + s) * 3;
  const float cx = ctr[0], cy = ctr[1], cz = ctr[2];
  const float* pb = xyz  + (size_t)b * NSRC * 3;
  const float* fb = feat + (size_t)b * NSRC * 3;
  const size_t nbase = (size_t)q * GRP;

  int cnt = 0;
  bool haveFirst = false;
  float f_rx = 0, f_ry = 0, f_rz = 0, f_f0 = 0, f_f1 = 0, f_f2 = 0;

  for (int c0 = 0; c0 < NSRC && cnt < GRP; c0 += 32) {
    const int p = c0 + lane;
    const float dx = pb[p * 3 + 0] - cx;
    const float dy = pb[p * 3 + 1] - cy;
    const float dz = pb[p * 3 + 2] - cz;
    const float d2 = dx * dx + dy * dy + dz * dz;
    const bool mine = (d2 <= RAD2);
    const unsigned m = __builtin_amdgcn_ballot_w32(mine);
    if (m) {
      const int ord = cnt + __popc(m & ((1u << lane) - 1u));
      if (mine && ord < GRP) {
        X0[(size_t)0 * NTOT + nbase + ord] = dx;
        X0[(size_t)1 * NTOT + nbase + ord] = dy;
        X0[(size_t)2 * NTOT + nbase + ord] = dz;
        X0[(size_t)3 * NTOT + nbase + ord] = fb[p * 3 + 0];
        X0[(size_t)4 * NTOT + nbase + ord] = fb[p * 3 + 1];
        X0[(size_t)5 * NTOT + nbase + ord] = fb[p * 3 + 2];
      }
      if (!haveFirst) {   // remember first in-range point for fill
        const int fl = __ffs(m) - 1;
        f_rx = __shfl(dx, fl, 32);
        f_ry = __shfl(dy, fl, 32);
        f_rz = __shfl(dz, fl, 32);
        const int fp = c0 + fl;
        f_f0 = fb[fp * 3 + 0]; f_f1 = fb[fp * 3 + 1]; f_f2 = fb[fp * 3 + 2];
        haveFirst = true;
      }
      cnt += __popc(m);
    }
  }
  if (!haveFirst) {  // degenerate (never happens: center is its own neighbor)
    const int p = NSRC - 1;
    f_rx = pb[p * 3 + 0] - cx; f_ry = pb[p * 3 + 1] - cy; f_rz = pb[p * 3 + 2] - cz;
    f_f0 = fb[p * 3 + 0]; f_f1 = fb[p * 3 + 1]; f_f2 = fb[p * 3 + 2];
    cnt = 0;
  }
  if (cnt < GRP && lane >= cnt) {    // pad remaining slots with first neighbor
    X0[(size_t)0 * NTOT + nbase + lane] = f_rx;
    X0[(size_t)1 * NTOT + nbase + lane] = f_ry;
    X0[(size_t)2 * NTOT + nbase + lane] = f_rz;
    X0[(size_t)3 * NTOT + nbase + lane] = f_f0;
    X0[(size_t)4 * NTOT + nbase + lane] = f_f1;
    X0[(size_t)5 * NTOT + nbase + lane] = f_f2;
  }
}

// ---------------------------------------------------------------------------
// GEMM via V_WMMA_F32_16X16X4_F32:  Y[M][NTOT] = W[M][K] @ act(X[K][NTOT]) + b
//
// Workgroup = 8 waves staging a [K][128] activation panel into LDS (async
// direct-to-LDS on gfx1250), so X is read from HBM exactly ONCE regardless
// of how many M-tiles consume it. BN+ReLU of the previous layer is applied
// in-place on the LDS panel once; all 8 waves then feed WMMAs from LDS
// (144-float row stride -> the two half-waves of a B-tile hit disjoint
// bank sets).
//
// A-layout (32b 16x4): lane l holds rows m0+(l&15); VGPR j holds K=(l>>4)*2+j.
// B mirrors it; C/D: VGPR v holds row m0+v+8*(l>>4), col n0+(l&15). Bias is
// preloaded in C; per-channel sum/sumsq for the next BN accumulate in regs
// and flush with one atomicAdd per channel per wave.
// ---------------------------------------------------------------------------
template <int K, int M, bool FUSE>
__global__ __launch_bounds__(256) void gemm_bn_kernel(const float* __restrict__ X,
                                                      const float* __restrict__ W,
                                                      const float* __restrict__ bias,
                                                      const float* __restrict__ coefIn,
                                                      float* __restrict__ Y,
                                                      float* __restrict__ statSum,
                                                      float* __restrict__ statSq) {
  constexpr int KT    = (K + 3) / 4;   // K tiles of 4
  constexpr int MT    = M / 16;        // M tiles of 16
  constexpr int NSTEP = 8 / MT;        // n-subtile stride per wave
  constexpr int LSTR  = 144;           // LDS row stride in floats
  constexpr int CH16  = K * 32;        // 16-byte chunks per panel
  constexpr int PITER = (CH16 + 255) / 256;
  __shared__ float panel[K * LSTR];

  const int t     = threadIdx.x;
  const int lane  = t & 31;
  const int wv    = t >> 5;
  const int mt    = wv % MT;
  const int nsub0 = wv / MT;
  const int m0    = mt * 16;
  const int half  = lane >> 4;
  const int nl    = lane & 15;

  // Preload the whole A (weight) panel for this wave's M tile into registers.
  float a0[KT], a1[KT];
#pragma unroll
  for (int kt = 0; kt < KT; ++kt) {
    const int c = kt * 4 + half * 2;
    a0[kt] = (c     < K) ? W[(m0 + nl) * K + c]     : 0.0f;
    a1[kt] = (c + 1 < K) ? W[(m0 + nl) * K + c + 1] : 0.0f;
  }
  v8f cinit;
#pragma unroll
  for (int v = 0; v < 8; ++v) cinit[v] = bias[m0 + v + half * 8];

  float sAcc[8], qAcc[8];
#pragma unroll
  for (int v = 0; v < 8; ++v) { sAcc[v] = 0.0f; qAcc[v] = 0.0f; }

  for (int chunk = blockIdx.x; chunk < NCHUNK; chunk += gridDim.x) {
    const size_t nbase = (size_t)chunk * 128;

    // ---- stage X[:, nbase..nbase+127] into LDS ----
#pragma unroll
    for (int i = 0; i < PITER; ++i) {
      const int ch = t + i * 256;
      if (ch < CH16) {
        const int row = ch >> 5;
        const int cg  = (ch & 31) * 4;
        const float* gp = X + (size_t)row * NTOT + nbase + cg;
        float* lp = &panel[row * LSTR + cg];
#if USE_ASYNC_LDS
        __builtin_amdgcn_global_load_async_to_lds_b128(
            (gv4i_p)const_cast<float*>(gp), (lv4i_p)lp, 0, 0);
#else
        *(float4*)lp = *(const float4*)gp;
#endif
      }
    }
#if USE_ASYNC_LDS
    wait_async0();
#endif
    __syncthreads();

    if constexpr (FUSE) {  // BN+ReLU of previous layer, once per element
#pragma unroll
      for (int i = 0; i < PITER; ++i) {
        const int ch = t + i * 256;
        if (ch < CH16) {
          const int row = ch >> 5;
          const int cg  = (ch & 31) * 4;
          const float sc = coefIn[row];
          const float sh = coefIn[K + row];
          float* lp = &panel[row * LSTR + cg];
          float4 v = *(float4*)lp;
          v.x = fmaxf(v.x * sc + sh, 0.0f);
          v.y = fmaxf(v.y * sc + sh, 0.0f);
          v.z = fmaxf(v.z * sc + sh, 0.0f);
          v.w = fmaxf(v.w * sc + sh, 0.0f);
          *(float4*)lp = v;
        }
      }
      __syncthreads();
    }

    // ---- compute: wave owns M tile mt, walks n-subtiles nsub0, +NSTEP ----
    for (int ns = nsub0; ns < 8; ns += NSTEP) {
      const int col = ns * 16 + nl;
      v8f acc = cinit;
#pragma unroll
      for (int kt = 0; kt < KT; ++kt) {
        const int c = kt * 4 + half * 2;
        float b0 = (c     < K) ? panel[c * LSTR + col]       : 0.0f;
        float b1 = (c + 1 < K) ? panel[(c + 1) * LSTR + col] : 0.0f;
        v2f av; av[0] = a0[kt]; av[1] = a1[kt];
        v2f bv; bv[0] = b0;     bv[1] = b1;
        acc = __builtin_amdgcn_wmma_f32_16x16x4_f32(false, av, false, bv,
                                                    (short)0, acc, false, false);
      }
      const size_t n = nbase + col;
#pragma unroll
      for (int v = 0; v < 8; ++v) {
        const float y = acc[v];
        Y[(size_t)(m0 + v + half * 8) * NTOT + n] = y;
        sAcc[v] += y;
        qAcc[v] += y * y;
      }
    }
    __syncthreads();   // everyone done reading before panel is overwritten
  }

  // Stats: reduce across the 16 lanes sharing each channel row, then one
  // atomicAdd per channel per wave.
#pragma unroll
  for (int v = 0; v < 8; ++v) {
#pragma unroll
    for (int off = 8; off >= 1; off >>= 1) {
      sAcc[v] += __shfl_xor(sAcc[v], off, 32);
      qAcc[v] += __shfl_xor(qAcc[v], off, 32);
    }
  }
  if (nl == 0) {
#pragma unroll
    for (int v = 0; v < 8; ++v) {
      atomicAdd(&statSum[m0 + v + half * 8], sAcc[v]);
      atomicAdd(&statSq [m0 + v + half * 8], qAcc[v]);
    }
  }
}

// ---------------------------------------------------------------------------
// Small helpers: zero the stat accumulators; fold (sum, sumsq) into the BN
// affine coefficients  y_hat = y*scale + shift  with scale = g*rsqrt(var+eps).
// ---------------------------------------------------------------------------
__global__ void zero_kernel(float* p, int n) {
  const int i = blockIdx.x * 256 + threadIdx.x;
  if (i < n) p[i] = 0.0f;
}

__global__ void bnfinal_kernel(const float* __restrict__ sum, const float* __restrict__ sq,
                               const float* __restrict__ g,   const float* __restrict__ beta,
                               float* __restrict__ coef, int M) {
  const int c = blockIdx.x * 64 + threadIdx.x;
  if (c < M) {
    const float inv = 1.0f / (float)NTOT;
    const float mu  = sum[c] * inv;
    const float var = sq[c] * inv - mu * mu;
    const float sc  = g[c] * rsqrtf(var + BN_EPS);
    coef[c]     = sc;
    coef[M + c] = beta[c] - mu * sc;
  }
}

// ---------------------------------------------------------------------------
// Final: BN2 + ReLU + max over the 32-point group + transpose to [B,S,128].
// Thread t handles channel o = t>>14 at position bs = t&16383; it reads its
// group as 8 x b128 loads (adjacent threads read adjacent 128B chunks).
// ---------------------------------------------------------------------------
__global__ __launch_bounds__(256) void maxpool_kernel(const float* __restrict__ Y3,
                                                      const float* __restrict__ coef,
                                                      float* __restrict__ out) {
  const int tid = blockIdx.x * 256 + threadIdx.x;   // 128 * 16384 threads
  const int o   = tid >> 14;
  const int bs  = tid & 16383;
  const float sc = coef[o], sh = coef[128 + o];
  const float4* p4 = (const float4*)(Y3 + (size_t)o * NTOT + (size_t)bs * GRP);
  float m = 0.0f;                                   // relu output is >= 0
#pragma unroll
  for (int g = 0; g < 8; ++g) {
    const float4 v = p4[g];
    m = fmaxf(m, fmaxf(v.x * sc + sh, 0.0f));
    m = fmaxf(m, fmaxf(v.y * sc + sh, 0.0f));
    m = fmaxf(m, fmaxf(v.z * sc + sh, 0.0f));
    m = fmaxf(m, fmaxf(v.w * sc + sh, 0.0f));
  }
  out[(size_t)bs * 128 + o] = m;
}

// ---------------------------------------------------------------------------
// Launch
// ---------------------------------------------------------------------------
extern "C" void kernel_launch(void* const* d_in, const int* in_sizes, int n_in,
                              void* d_out, int out_size, void* d_ws, size_t ws_size,
                              hipStream_t stream) {
  const float* xyz  = (const float*)d_in[0];
  const float* feat = (const float*)d_in[1];
  const float* w0 = (const float*)d_in[2];
  const float* b0 = (const float*)d_in[3];
  const float* g0 = (const float*)d_in[4];
  const float* t0 = (const float*)d_in[5];
  const float* w1 = (const float*)d_in[6];
  const float* b1 = (const float*)d_in[7];
  const float* g1 = (const float*)d_in[8];
  const float* t1 = (const float*)d_in[9];
  const float* w2 = (const float*)d_in[10];
  const float* b2 = (const float*)d_in[11];
  const float* g2 = (const float*)d_in[12];
  const float* t2 = (const float*)d_in[13];

  float* out_xyz  = (float*)d_out;                       // [16,1024,3]
  float* out_feat = out_xyz + (size_t)BATCH * NPTS * 3;  // [16,1024,128]

  // Workspace layout (f32):
  //   X0   : 6   * NTOT   (grouped features, channel-major)
  //   bufA : 128 * NTOT   (rows 0..63 = Y1; later reused in full as Y3)
  //   Y2   : 64  * NTOT
  //   stats: 1024 floats  (sums, sumsqs, BN coefs)
  float* X0   = (float*)d_ws;
  float* bufA = X0 + (size_t)6 * NTOT;
  float* Y2   = bufA + (size_t)128 * NTOT;
  float* st   = Y2 + (size_t)64 * NTOT;
  float* sum0 = st;        float* sq0 = st + 64;
  float* sum1 = st + 128;  float* sq1 = st + 192;
  float* sum2 = st + 256;  float* sq2 = st + 384;
  float* coef0 = st + 512;   // [scale64 | shift64]
  float* coef1 = st + 640;
  float* coef2 = st + 768;   // [scale128 | shift128]

  // 1. farthest point sampling -> new_xyz (in d_out)
  fps_kernel<<<BATCH, 1024, 0, stream>>>(xyz, out_xyz);

  // 2. ball query + gather + concat -> X0
  ballquery_kernel<<<(BATCH * NPTS) / 8, 256, 0, stream>>>(xyz, feat, out_xyz, X0);

  // 3. zero stat accumulators
  zero_kernel<<<4, 256, 0, stream>>>(st, 1024);

  // 4. layer 0: 6 -> 64 (WMMA), stats for BN0
  gemm_bn_kernel<6, 64, false><<<512, 256, 0, stream>>>(X0, w0, b0, nullptr,
                                                        bufA, sum0, sq0);
  bnfinal_kernel<<<1, 64, 0, stream>>>(sum0, sq0, g0, t0, coef0, 64);

  // 5. layer 1: 64 -> 64, BN0+ReLU applied once on the LDS panel
  gemm_bn_kernel<64, 64, true><<<512, 256, 0, stream>>>(bufA, w1, b1, coef0,
                                                        Y2, sum1, sq1);
  bnfinal_kernel<<<1, 64, 0, stream>>>(sum1, sq1, g1, t1, coef1, 64);

  // 6. layer 2: 64 -> 128, BN1+ReLU fused; Y3 reuses bufA (Y1 is dead)
  gemm_bn_kernel<64, 128, true><<<512, 256, 0, stream>>>(Y2, w2, b2, coef1,
                                                         bufA, sum2, sq2);
  bnfinal_kernel<<<2, 64, 0, stream>>>(sum2, sq2, g2, t2, coef2, 128);

  // 7. BN2 + ReLU + group max + transpose -> d_out features
  maxpool_kernel<<<(128 * BATCH * NPTS) / 256, 256, 0, stream>>>(bufA, coef2, out_feat);
}